// SimpleGATv2Net_8564164788936
// MI455X (gfx1250) — compile-verified
//
#include <hip/hip_runtime.h>
#include <hip/hip_bf16.h>

typedef float v2f __attribute__((ext_vector_type(2)));
typedef float v8f __attribute__((ext_vector_type(8)));

#define NEG_SLOPE 0.2f

__device__ __forceinline__ int ord_f32(float f) {
    int i = __float_as_int(f);
    return i >= 0 ? i : (i ^ 0x7fffffff);
}
__device__ __forceinline__ float unord_f32(int i) {
    return __int_as_float(i >= 0 ? i : (i ^ 0x7fffffff));
}

// ---------------------------------------------------------------------------
// Init: out = bias (broadcast), m_ord = -inf (ordered-int min), denom = 0
// ---------------------------------------------------------------------------
__global__ void gat_init(float* __restrict__ out, const float* __restrict__ bias,
                         int* __restrict__ m_ord, float* __restrict__ denom, int N) {
    int tid = blockIdx.x * blockDim.x + threadIdx.x;
    long total = (long)N * 64;
    if (tid < total) out[tid] = bias[tid & 63];
    if (tid < N) { m_ord[tid] = (int)0x80000000; denom[tid] = 0.0f; }
}

// ---------------------------------------------------------------------------
// One K-step (K+=4): 4x V_WMMA_F32_16X16X4_F32 across the 4 column tiles.
// ---------------------------------------------------------------------------
__device__ __forceinline__ void wmma_step(v8f acc[4], v2f a,
                                          const float* __restrict__ W,
                                          int ka, int lm) {
#pragma unroll
    for (int t = 0; t < 4; ++t) {
        int col = t * 16 + lm;
        v2f b;
        b.x = W[(size_t)ka * 64 + col];
        b.y = W[(size_t)(ka + 1) * 64 + col];
        acc[t] = __builtin_amdgcn_wmma_f32_16x16x4_f32(
            false, a, false, b, (short)0, acc[t], false, false);
    }
}

// ---------------------------------------------------------------------------
// GEMM: each wave computes a 16x64 tile of ONE output (wave parity selects
// xl/Wl vs xr/Wr) -> 32 accumulator VGPRs, no spills, high occupancy.
// Main loop has an EVEN trip count (30) so the x2 unroll has no mid-body
// exit (kills the accumulator mov chain); last two K-steps are peeled and
// the x_binary (col 127) select is a branchless v_cndmask.
// ---------------------------------------------------------------------------
__global__ void gat_gemm(const float* __restrict__ xf, const float* __restrict__ xb,
                         const float* __restrict__ Wl, const float* __restrict__ bl,
                         const float* __restrict__ Wr, const float* __restrict__ br,
                         float* __restrict__ xl, float* __restrict__ xr, int N) {
    const int FIN = 128, C = 64;
    int wid  = (int)((blockIdx.x * blockDim.x + threadIdx.x) >> 5);
    int which = wid & 1;           // 0 -> (Wl,bl,xl), 1 -> (Wr,br,xr)
    int wave  = wid >> 1;
    int lane = threadIdx.x & 31;
    int half = lane >> 4;          // selects K pair (A/B) and M group (C/D)
    int lm   = lane & 15;
    int node_base = wave * 16;
    if (node_base >= N) return;    // wave-uniform exit, EXEC stays full for WMMA

    const float* W    = which ? Wr : Wl;
    const float* bvec = which ? br : bl;
    float*       dst  = which ? xr : xl;

    int row = node_base + lm;
    if (row >= N) row = N - 1;     // clamp (stores are guarded below)
    const float* xrow = xf + (size_t)row * (FIN - 1);

    v8f acc[4] = {};

    // K = 0..119: 30 iterations (even) -> clean x2 unroll, no mid-body exit.
    for (int k0 = 0; k0 < FIN - 8; k0 += 4) {
        int ka = k0 + 2 * half;
        v2f a;
        a.x = xrow[ka];
        a.y = xrow[ka + 1];
        wmma_step(acc, a, W, ka, lm);
    }
    // Peeled step k0 = 120: still pure x_float.
    {
        int ka = (FIN - 8) + 2 * half;     // 120 or 122
        v2f a;
        a.x = xrow[ka];
        a.y = xrow[ka + 1];
        wmma_step(acc, a, W, ka, lm);
    }
    // Peeled last step k0 = 124: column 127 = x_binary, branchless select.
    {
        int ka = (FIN - 4) + 2 * half;     // 124 or 126
        float x_lo = xrow[ka];             // cols 124 / 126 (always x_float)
        float x125 = xrow[125];
        float xbin = xb[row];
        v2f a;
        a.x = x_lo;
        a.y = (half == 0) ? x125 : xbin;   // v_cndmask, no EXEC change
        wmma_step(acc, a, W, ka, lm);
    }

    if (node_base + 16 <= N) {
        // Full tile: unguarded stores.
#pragma unroll
        for (int t = 0; t < 4; ++t) {
            int col = t * 16 + lm;
#pragma unroll
            for (int r = 0; r < 8; ++r) {
                int orow = node_base + r + 8 * half;   // C/D layout: M = r + 8*half
                dst[(size_t)orow * C + col] = acc[t][r] + bvec[col];
            }
        }
    } else {
#pragma unroll
        for (int t = 0; t < 4; ++t) {
            int col = t * 16 + lm;
#pragma unroll
            for (int r = 0; r < 8; ++r) {
                int orow = node_base + r + 8 * half;
                if (orow < N)
                    dst[(size_t)orow * C + col] = acc[t][r] + bvec[col];
            }
        }
    }
}

// ---------------------------------------------------------------------------
// Per-edge logits: a . LeakyReLU(xl[src] + xr[dst]); one wave per edge,
// lanes cover channels {l, l+32}. Also segment-max via ordered-int atomicMax.
// ---------------------------------------------------------------------------
__global__ void gat_logits(const int* __restrict__ ei, const float* __restrict__ xl,
                           const float* __restrict__ xr, const float* __restrict__ att,
                           float* __restrict__ logits, int* __restrict__ m_ord,
                           int E, int Etot) {
    int gid = blockIdx.x * blockDim.x + threadIdx.x;
    int e = gid >> 5;
    int lane = threadIdx.x & 31;
    if (e >= Etot) return;
    int src, dst;
    if (e < E) { src = ei[e]; dst = ei[E + e]; } else { src = dst = e - E; }
    const float* pl = xl + (size_t)src * 64;
    const float* pr = xr + (size_t)dst * 64;
    float z0 = pl[lane]      + pr[lane];
    float z1 = pl[lane + 32] + pr[lane + 32];
    z0 = z0 >= 0.f ? z0 : NEG_SLOPE * z0;
    z1 = z1 >= 0.f ? z1 : NEG_SLOPE * z1;
    float p = att[lane] * z0 + att[lane + 32] * z1;
#pragma unroll
    for (int off = 16; off; off >>= 1) p += __shfl_down(p, off, 32);
    if (lane == 0) {
        logits[e] = p;
        atomicMax(m_ord + dst, ord_f32(p));
    }
}

// ---------------------------------------------------------------------------
// ex = exp(logit - max); denom[dst] += ex.  (logits overwritten in place)
// ---------------------------------------------------------------------------
__global__ void gat_softmax(const int* __restrict__ ei, float* __restrict__ logits,
                            const int* __restrict__ m_ord, float* __restrict__ denom,
                            int E, int Etot) {
    int e = blockIdx.x * blockDim.x + threadIdx.x;
    if (e >= Etot) return;
    int dst = (e < E) ? ei[E + e] : (e - E);
    float ex = expf(logits[e] - unord_f32(m_ord[dst]));
    logits[e] = ex;
    atomicAdd(denom + dst, ex);
}

// ---------------------------------------------------------------------------
// out[dst] += (ex/denom[dst]) * xl[src]; one wave per edge, 2 atomics/lane.
// out (25.6 MB) and xl (25.6 MB) are L2-resident -> L2-rate atomics.
// ---------------------------------------------------------------------------
__global__ void gat_aggregate(const int* __restrict__ ei, const float* __restrict__ xl,
                              const float* __restrict__ logits,
                              const float* __restrict__ denom,
                              float* __restrict__ out, int E, int Etot) {
    int gid = blockIdx.x * blockDim.x + threadIdx.x;
    int e = gid >> 5;
    int lane = threadIdx.x & 31;
    if (e >= Etot) return;
    int src, dst;
    if (e < E) { src = ei[e]; dst = ei[E + e]; } else { src = dst = e - E; }
    float alpha = logits[e] / denom[dst];
    const float* pl = xl + (size_t)src * 64;
    float* po = out + (size_t)dst * 64;
    atomicAdd(po + lane,      alpha * pl[lane]);
    atomicAdd(po + lane + 32, alpha * pl[lane + 32]);
}

// ---------------------------------------------------------------------------
extern "C" void kernel_launch(void* const* d_in, const int* in_sizes, int n_in,
                              void* d_out, int out_size, void* d_ws, size_t ws_size,
                              hipStream_t stream) {
    const float* xf   = (const float*)d_in[0];   // [N, 127]
    const float* xb   = (const float*)d_in[1];   // [N]
    const int*   ei   = (const int*)  d_in[2];   // [2, E]
    const float* Wl   = (const float*)d_in[3];   // [128, 64]
    const float* bl   = (const float*)d_in[4];   // [64]
    const float* Wr   = (const float*)d_in[5];   // [128, 64]
    const float* br   = (const float*)d_in[6];   // [64]
    const float* att  = (const float*)d_in[7];   // [64]
    const float* bias = (const float*)d_in[8];   // [64]

    int N    = in_sizes[1];
    int E    = in_sizes[2] / 2;
    int Etot = E + N;                 // self-loops appended

    float* out    = (float*)d_out;
    float* xl     = (float*)d_ws;
    float* xr     = xl + (size_t)N * 64;
    float* logits = xr + (size_t)N * 64;
    int*   m_ord  = (int*)(logits + Etot);
    float* denom  = (float*)(m_ord + N);

    long initElems = (long)N * 64;
    gat_init<<<(unsigned)((initElems + 255) / 256), 256, 0, stream>>>(out, bias, m_ord, denom, N);

    int nwaves = 2 * ((N + 15) / 16);   // 2 waves (xl, xr) per 16-row tile
    gat_gemm<<<(nwaves + 7) / 8, 256, 0, stream>>>(xf, xb, Wl, bl, Wr, br, xl, xr, N);

    gat_logits<<<(Etot + 7) / 8, 256, 0, stream>>>(ei, xl, xr, att, logits, m_ord, E, Etot);
    gat_softmax<<<(Etot + 255) / 256, 256, 0, stream>>>(ei, logits, m_ord, denom, E, Etot);
    gat_aggregate<<<(Etot + 7) / 8, 256, 0, stream>>>(ei, xl, logits, denom, out, E, Etot);
}